// DynamicGraphEmbedding_37048387895632
// MI455X (gfx1250) — compile-verified
//
#include <hip/hip_runtime.h>
#include <hip/hip_bf16.h>

typedef __attribute__((ext_vector_type(2))) float v2f;
typedef __attribute__((ext_vector_type(8))) float v8f;

#define Nn 512
#define Ss 256
#define Bb 64
#define K1 (Bb * Ss)          // 16384
#define TOPP 154              // k+1, k = int(512*0.3) = 153
#define EPSV 1e-12f
#define ZSIM 8                // split-K factor for mean_sim GEMM
#define ZCOM 4                // split-K factor for common GEMM

// ---- workspace float offsets (with reuse) ----
// buf0: Xraw [N, B*S]  -> later reused as Out_nbs [N, B*S]
// buf1: Xn   [N, B*S]  -> later reused as XW      [N*B, S] == [N, B*S]
static constexpr size_t OFF_XRAW = 0;
static constexpr size_t OFF_XN   = OFF_XRAW + (size_t)Nn * K1;
static constexpr size_t OFF_SIM  = OFF_XN   + (size_t)Nn * K1;
static constexpr size_t OFF_EM   = OFF_SIM  + (size_t)Nn * Nn;
static constexpr size_t OFF_ADJ  = OFF_EM   + (size_t)Nn * Nn;
static constexpr size_t OFF_NB   = OFF_ADJ  + (size_t)Nn * Nn;
static constexpr size_t OFF_COM  = OFF_NB   + (size_t)Nn * Nn;
static constexpr size_t OFF_WM   = OFF_COM  + (size_t)Nn * Nn;
static constexpr size_t OFF_AM   = OFF_WM   + (size_t)Nn * Nn;
static constexpr size_t OFF_INVN = OFF_AM   + (size_t)Nn * Nn;
static constexpr size_t OFF_DEG  = OFF_INVN + (size_t)Nn * Bb;
static constexpr size_t OFF_MAXC = OFF_DEG  + Nn;
static constexpr size_t OFF_WT   = OFF_MAXC + 8;
static constexpr size_t OFF_PART = OFF_WT   + (size_t)Ss * Ss;   // [ZSIM][N][N]
// total = OFF_PART + ZSIM*N*N ~= 20.9M floats (~84 MB)

// ---------------------------------------------------------------------------
// inverse L2 norm per (n,b):  invn[b*N + n] = 1/max(||x[b,:,n]||, eps)
__global__ __launch_bounds__(256) void k_invnorm(const float* __restrict__ x,
                                                 float* __restrict__ invn) {
  int flat = blockIdx.x * 256 + threadIdx.x;      // N*B = 32768 total
  int b = flat >> 9;
  int n = flat & (Nn - 1);
  const float* p = x + (size_t)b * Ss * Nn + n;
  float ss = 0.f;
  for (int s = 0; s < Ss; ++s) { float v = p[(size_t)s * Nn]; ss += v * v; }
  float nrm = fmaxf(sqrtf(ss), EPSV);
  invn[flat] = 1.0f / nrm;                        // layout [b][n], coalesced
}

// transpose x[b,s,n] -> Xraw[n, b*S+s] and Xn = Xraw * invnorm
__global__ __launch_bounds__(256) void k_prep_x(const float* __restrict__ x,
                                                const float* __restrict__ invn,
                                                float* __restrict__ Xraw,
                                                float* __restrict__ Xn) {
  __shared__ float tile[32][33];
  int n0 = blockIdx.x * 32, s0 = blockIdx.y * 32, b = blockIdx.z;
  int tx = threadIdx.x, ty = threadIdx.y;
#pragma unroll
  for (int r = 0; r < 4; ++r) {
    int sl = ty + 8 * r;
    tile[sl][tx] = x[((size_t)b * Ss + (s0 + sl)) * Nn + n0 + tx];
  }
  __syncthreads();
#pragma unroll
  for (int r = 0; r < 4; ++r) {
    int nl = ty + 8 * r;
    int n = n0 + nl;
    float v = tile[tx][nl];                       // element (n, s0+tx)
    size_t off = (size_t)n * K1 + (size_t)b * Ss + s0 + tx;
    Xraw[off] = v;
    Xn[off]   = v * invn[b * Nn + n];
  }
}

// WT[t][s] = W[s][t]
__global__ __launch_bounds__(256) void k_wt(const float* __restrict__ W,
                                            float* __restrict__ WT) {
  int idx = blockIdx.x * 256 + threadIdx.x;       // S*S
  int s = idx >> 8, t = idx & (Ss - 1);
  WT[(size_t)t * Ss + s] = W[idx];
}

// ---------------------------------------------------------------------------
// Generic fp32 WMMA GEMM with optional split-K over blockIdx.z.
//   BT=true : C[i,j] = alpha * sum_k Am[i,k] * Bm[j,k]   (Bm row-major [Nc,K])
//   BT=false: C[i,j] = alpha * sum_k Am[i,k] * Bm[k,j]   (Bm row-major [K,Nc])
// K is the per-z chunk length. blockIdx.z == z writes its partial to
//   Cm + z * (gridDim.x*128) * ldc   (z-slices contiguous; gridDim.z==1 => Cm).
// Block = 256 threads = 8 waves (4x2). Wave tile 32x32 = 2x2 WMMA 16x16x4.
// Grid: (M/128, Nc/64, Z). All dims multiples of tile sizes.
__device__ __forceinline__ v8f wmma4(v2f a, v2f b, v8f c) {
  return __builtin_amdgcn_wmma_f32_16x16x4_f32(false, a, false, b, (short)0, c,
                                               false, false);
}

template <bool BT>
__global__ __launch_bounds__(256) void gemm_wmma(const float* __restrict__ Am,
                                                 const float* __restrict__ Bm,
                                                 float* __restrict__ Cm, int K,
                                                 int lda, int ldb, int ldc,
                                                 float alpha) {
  const int tid = threadIdx.x;
  const int wave = tid >> 5;
  const int lane = tid & 31;
  const int half = lane >> 4;          // K-half selector
  const int lm = lane & 15;            // row (A) / col (B,C) within 16
  const int wm = wave >> 1;            // 0..3
  const int wn = wave & 1;             // 0..1
  const int i0 = blockIdx.x * 128 + wm * 32;
  const int j0 = blockIdx.y * 64 + wn * 32;

  // split-K offsets
  const size_t koff = (size_t)blockIdx.z * K;
  Am += koff;                                   // within-row K offset
  if (BT) Bm += koff; else Bm += koff * ldb;    // rows for NN layout
  float* Cz = Cm + (size_t)blockIdx.z * (size_t)gridDim.x * 128 * ldc;

  v8f acc[2][2] = {};

  const float* a0p = Am + (size_t)(i0 + lm) * lda;
  const float* a1p = Am + (size_t)(i0 + 16 + lm) * lda;
  const float* b0p; const float* b1p;
  if (BT) {
    b0p = Bm + (size_t)(j0 + lm) * ldb;
    b1p = Bm + (size_t)(j0 + 16 + lm) * ldb;
  } else {
    b0p = Bm + (j0 + lm);
    b1p = Bm + (j0 + 16 + lm);
  }

  for (int k = 0; k < K; k += 4) {
    const int kk = k + 2 * half;
    v2f a0 = *(const v2f*)(a0p + kk);
    v2f a1 = *(const v2f*)(a1p + kk);
    v2f b0, b1;
    if (BT) {
      b0 = *(const v2f*)(b0p + kk);
      b1 = *(const v2f*)(b1p + kk);
    } else {
      b0.x = b0p[(size_t)kk * ldb];
      b0.y = b0p[(size_t)(kk + 1) * ldb];
      b1.x = b1p[(size_t)kk * ldb];
      b1.y = b1p[(size_t)(kk + 1) * ldb];
    }
    acc[0][0] = wmma4(a0, b0, acc[0][0]);
    acc[0][1] = wmma4(a0, b1, acc[0][1]);
    acc[1][0] = wmma4(a1, b0, acc[1][0]);
    acc[1][1] = wmma4(a1, b1, acc[1][1]);
  }

#pragma unroll
  for (int ti = 0; ti < 2; ++ti)
#pragma unroll
    for (int tj = 0; tj < 2; ++tj)
#pragma unroll
      for (int v = 0; v < 8; ++v) {
        int row = i0 + ti * 16 + 8 * half + v;
        int col = j0 + tj * 16 + lm;
        Cz[(size_t)row * ldc + col] = alpha * acc[ti][tj][v];
      }
}

// deterministic fixed-order reduction of Z split-K partials (each [N][N])
__global__ __launch_bounds__(256) void k_reduceZ(const float* __restrict__ part,
                                                 float* __restrict__ outm,
                                                 int Z, float alpha) {
  size_t idx = (size_t)blockIdx.x * 256 + threadIdx.x;   // N*N total
  float s = 0.f;
  for (int z = 0; z < Z; ++z) s += part[(size_t)z * Nn * Nn + idx];
  outm[idx] = alpha * s;
}

// ---------------------------------------------------------------------------
// per-row exact top-(k+1) via bitonic sort of 512 (value,index) keys;
// drop self (or last if self absent); write float mask row.
__global__ __launch_bounds__(256) void k_topk(const float* __restrict__ sim,
                                              float* __restrict__ em) {
  __shared__ unsigned long long keys[Nn];
  __shared__ int selfpos;
  const int i = blockIdx.x, tid = threadIdx.x;
  for (int j = tid; j < Nn; j += 256) {
    unsigned u = __float_as_uint(sim[(size_t)i * Nn + j]);
    unsigned m = (u & 0x80000000u) ? ~u : (u | 0x80000000u);  // order-preserving
    keys[j] = ((unsigned long long)(~m) << 32) | (unsigned)j; // asc => desc value, asc idx
  }
  if (tid == 0) selfpos = -1;
  __syncthreads();
  for (int kk = 2; kk <= Nn; kk <<= 1)
    for (int jj = kk >> 1; jj > 0; jj >>= 1) {
      for (int idx = tid; idx < Nn; idx += 256) {
        int l = idx ^ jj;
        if (l > idx) {
          unsigned long long a = keys[idx], b = keys[l];
          bool asc = ((idx & kk) == 0);
          if ((a > b) == asc) { keys[idx] = b; keys[l] = a; }
        }
      }
      __syncthreads();
    }
  for (int p = tid; p < TOPP; p += 256)
    if ((unsigned)(keys[p] & 0xffffffffu) == (unsigned)i) selfpos = p;
  __syncthreads();
  const int drop = (selfpos >= 0) ? selfpos : (TOPP - 1);
  for (int j = tid; j < Nn; j += 256) em[(size_t)i * Nn + j] = 0.0f;
  __syncthreads();
  for (int p = tid; p < TOPP; p += 256)
    if (p != drop)
      em[(size_t)i * Nn + (keys[p] & 0xffffffffu)] = 1.0f;
}

// adj = sym(em) w/o diag; nb = adj + I
__global__ __launch_bounds__(256) void k_adj_nb(const float* __restrict__ em,
                                                float* __restrict__ adj,
                                                float* __restrict__ nb) {
  int idx = blockIdx.x * 256 + threadIdx.x;       // N*N
  int i = idx >> 9, j = idx & (Nn - 1);
  bool e = (em[idx] != 0.f) || (em[(size_t)j * Nn + i] != 0.f);
  float a = (i != j && e) ? 1.f : 0.f;
  adj[idx] = a;
  nb[idx] = (i == j) ? 1.f : a;
}

__global__ void k_maxinit(unsigned* __restrict__ mx) { *mx = 0u; }

__global__ __launch_bounds__(256) void k_maxred(const float* __restrict__ src,
                                                unsigned* __restrict__ mx) {
  __shared__ float red[256];
  int tid = threadIdx.x;
  red[tid] = src[blockIdx.x * 256 + tid];          // all values >= 0
  __syncthreads();
  for (int s = 128; s > 0; s >>= 1) {
    if (tid < s) red[tid] = fmaxf(red[tid], red[tid + s]);
    __syncthreads();
  }
  if (tid == 0) atomicMax(mx, __float_as_uint(red[0]));
}

// Wmat[i][j] = (sim[j,i] + struct[j,i]) * em[j,i];  deg[i] = sum_j Wmat[i][j]
__global__ __launch_bounds__(256) void k_we_deg(const float* __restrict__ sim,
                                                const float* __restrict__ em,
                                                const float* __restrict__ adj,
                                                const float* __restrict__ com,
                                                const float* __restrict__ maxc,
                                                float* __restrict__ Wmat,
                                                float* __restrict__ deg) {
  __shared__ float red[256];
  const int i = blockIdx.x, tid = threadIdx.x;
  const float mc = fmaxf(*maxc, EPSV);
  float sum = 0.f;
  for (int j = tid; j < Nn; j += 256) {
    size_t ji = (size_t)j * Nn + i;
    float c = com[ji];
    float st = (adj[ji] > 0.f && c > 1.0f) ? (c / mc) * c : 0.f;
    float we = (sim[ji] + st) * em[ji];
    Wmat[(size_t)i * Nn + j] = we;
    sum += we;
  }
  red[tid] = sum;
  __syncthreads();
  for (int s = 128; s > 0; s >>= 1) {
    if (tid < s) red[tid] += red[tid + s];
    __syncthreads();
  }
  if (tid == 0) deg[i] = red[0];
}

// A[i][j] = dinv(deg[i]) * Wmat[i][j] * dinv(deg[j])
__global__ __launch_bounds__(256) void k_buildA(const float* __restrict__ Wmat,
                                                const float* __restrict__ deg,
                                                float* __restrict__ A) {
  int idx = blockIdx.x * 256 + threadIdx.x;
  int i = idx >> 9, j = idx & (Nn - 1);
  float di = deg[i], dj = deg[j];
  float ii = (di == 0.f) ? 0.f : 1.0f / sqrtf(di);  // NaN for di<0, like deg**-0.5
  float ij = (dj == 0.f) ? 0.f : 1.0f / sqrtf(dj);
  A[idx] = ii * Wmat[idx] * ij;
}

// d_out[b,s,n] = Onbs[n, b*S+s] + bias[s]
__global__ __launch_bounds__(256) void k_out(const float* __restrict__ Onbs,
                                             const float* __restrict__ bias,
                                             float* __restrict__ out) {
  __shared__ float tile[32][33];
  int n0 = blockIdx.x * 32, s0 = blockIdx.y * 32, b = blockIdx.z;
  int tx = threadIdx.x, ty = threadIdx.y;
#pragma unroll
  for (int r = 0; r < 4; ++r) {
    int nl = ty + 8 * r;
    tile[nl][tx] = Onbs[(size_t)(n0 + nl) * K1 + (size_t)b * Ss + s0 + tx];
  }
  __syncthreads();
#pragma unroll
  for (int r = 0; r < 4; ++r) {
    int sl = ty + 8 * r;
    out[((size_t)b * Ss + (s0 + sl)) * Nn + n0 + tx] = tile[tx][sl] + bias[s0 + sl];
  }
}

// ---------------------------------------------------------------------------
extern "C" void kernel_launch(void* const* d_in, const int* in_sizes, int n_in,
                              void* d_out, int out_size, void* d_ws,
                              size_t ws_size, hipStream_t stream) {
  (void)in_sizes; (void)n_in; (void)out_size; (void)ws_size;
  const float* x    = (const float*)d_in[0];   // [B,S,N]
  const float* W    = (const float*)d_in[1];   // [S,S]
  const float* bias = (const float*)d_in[2];   // [S]
  float* ws  = (float*)d_ws;
  float* out = (float*)d_out;

  float* Xraw = ws + OFF_XRAW;                 // later: Out_nbs
  float* Xn   = ws + OFF_XN;                   // later: XW
  float* sim  = ws + OFF_SIM;
  float* em   = ws + OFF_EM;
  float* adj  = ws + OFF_ADJ;
  float* nb   = ws + OFF_NB;
  float* com  = ws + OFF_COM;
  float* Wm   = ws + OFF_WM;
  float* Am   = ws + OFF_AM;
  float* invn = ws + OFF_INVN;
  float* deg  = ws + OFF_DEG;
  float* maxc = ws + OFF_MAXC;
  float* WT   = ws + OFF_WT;
  float* part = ws + OFF_PART;                 // split-K partials

  // 1) norms + transposed (raw, normalized) copies
  k_invnorm<<<(Nn * Bb) / 256, 256, 0, stream>>>(x, invn);
  k_prep_x<<<dim3(Nn / 32, Ss / 32, Bb), dim3(32, 8), 0, stream>>>(x, invn, Xraw, Xn);
  k_wt<<<(Ss * Ss) / 256, 256, 0, stream>>>(W, WT);

  // 2) mean_sim = (1/B) * Xn @ Xn^T  [512 x 512, K = 16384], 8-way split-K
  gemm_wmma<true><<<dim3(Nn / 128, Nn / 64, ZSIM), 256, 0, stream>>>(
      Xn, Xn, part, K1 / ZSIM, K1, K1, Nn, 1.0f);
  k_reduceZ<<<(Nn * Nn) / 256, 256, 0, stream>>>(part, sim, ZSIM,
                                                 1.0f / (float)Bb);

  // 3) top-k mask, adjacency
  k_topk<<<Nn, 256, 0, stream>>>(sim, em);
  k_adj_nb<<<(Nn * Nn) / 256, 256, 0, stream>>>(em, adj, nb);

  // 4) common = nb @ nb^T  [512 x 512, K = 512], 4-way split-K (reuse partials)
  gemm_wmma<true><<<dim3(Nn / 128, Nn / 64, ZCOM), 256, 0, stream>>>(
      nb, nb, part, Nn / ZCOM, Nn, Nn, Nn, 1.0f);
  k_reduceZ<<<(Nn * Nn) / 256, 256, 0, stream>>>(part, com, ZCOM, 1.0f);
  k_maxinit<<<1, 1, 0, stream>>>((unsigned*)maxc);
  k_maxred<<<(Nn * Nn) / 256, 256, 0, stream>>>(com, (unsigned*)maxc);

  // 5) xw = x_raw @ W  as [N*B, S] x [S, S]^T-form (B operand = W^T), into Xn buf
  gemm_wmma<true><<<dim3((Nn * Bb) / 128, Ss / 64, 1), 256, 0, stream>>>(
      Xraw, WT, Xn, Ss, Ss, Ss, Ss, 1.0f);

  // 6) fused edge weights, degrees, propagation matrix A
  k_we_deg<<<Nn, 256, 0, stream>>>(sim, em, adj, com, maxc, Wm, deg);
  k_buildA<<<(Nn * Nn) / 256, 256, 0, stream>>>(Wm, deg, Am);

  // 7) Out_nbs = A @ XW  [512 x 16384, K = 512], into Xraw buf
  gemm_wmma<false><<<dim3(Nn / 128, K1 / 64, 1), 256, 0, stream>>>(
      Am, Xn, Xraw, Nn, Nn, K1, K1, 1.0f);

  // 8) transpose + bias -> [B, S, N]
  k_out<<<dim3(Nn / 32, Ss / 32, Bb), dim3(32, 8), 0, stream>>>(Xraw, bias, out);
}